// HierarchicalRouter_52544629899283
// MI455X (gfx1250) — compile-verified
//
#include <hip/hip_runtime.h>

typedef __attribute__((ext_vector_type(2))) float v2f;
typedef __attribute__((ext_vector_type(8))) float v8f;

#define DIM     1024
#define NEXP    64
#define NGRP    4
#define NTOK    32768          // B*S = 8*4096
#define NTILES  5              // 5 x 16 = 80 output cols (64 expert + 4 group + 12 pad)
#define NCOLS   80
#define TM      128            // tokens per block
#define TK      32             // K chunk
#define SA_STRIDE 36           // padded row stride for x tile in LDS (floats)
#define PAIR_STRIDE (NCOLS * 2)  // 160 floats per K-pair row in Wt-pair layout
#define NEGF    (-3.4028234663852886e38f)

// ---- CDNA5 async memory->LDS copy (ASYNCcnt-tracked, bypasses VGPRs) --------
__device__ __forceinline__ void async_ld_b128(uint32_t lds_addr, uint64_t gaddr) {
    asm volatile("global_load_async_to_lds_b128 %0, %1, off"
                 :: "v"(lds_addr), "v"(gaddr)
                 : "memory");
}
__device__ __forceinline__ void wait_async0() {
    asm volatile("s_wait_asynccnt 0" ::: "memory");
}
__device__ __forceinline__ uint32_t lds_off(const void* p) {
    // generic LDS address truncates to the 32-bit LDS byte offset (ISA 10.2)
    return (uint32_t)(uintptr_t)p;
}

// ---------------- prep: build K-pair-interleaved W^T in ws, zero accumulators ----
// layout: Wtp[(k>>1)*160 + n*2 + (k&1)]  -> each B fragment {K,K+1} is contiguous
__global__ void hr_prep_kernel(const float* __restrict__ Wg,
                               const float* __restrict__ We,
                               float* __restrict__ ws) {
    int i = blockIdx.x * 256 + threadIdx.x;
    if (i < DIM * NCOLS) {
        int k = i / NCOLS;
        int n = i - k * NCOLS;
        float v = 0.0f;
        if (n < NEXP)             v = We[(size_t)n * DIM + k];
        else if (n < NEXP + NGRP) v = Wg[(size_t)(n - NEXP) * DIM + k];
        ws[(size_t)(k >> 1) * PAIR_STRIDE + n * 2 + (k & 1)] = v;
    }
    if (i < 128) ws[(size_t)DIM * NCOLS + i] = 0.0f;  // imp[64] + cnt[64]
}

// ---------------- main fused router kernel -----------------------------------
__global__ __launch_bounds__(256) void hr_router_kernel(
        const float* __restrict__ x,      // [NTOK, DIM]
        const float* __restrict__ Wtp,    // pair-interleaved W^T, DIM*NCOLS floats
        int*   __restrict__ out_idx,      // [NTOK, 2]
        float* __restrict__ out_scores,   // [NTOK, 2]
        float* __restrict__ out_probs,    // [NTOK, NEXP]
        float* __restrict__ ws_imp,       // [NEXP]
        float* __restrict__ ws_cnt) {     // [NEXP]
    __shared__ float sA[2][TM * SA_STRIDE];
    __shared__ float sB[2][TK * NCOLS];
    __shared__ float sL[TM * (NCOLS + 1)];
    __shared__ int   sCnt[NEXP];
    const int SL_STRIDE = NCOLS + 1;      // 81, odd -> conflict-free column sums

    const int tid  = threadIdx.x;
    const int wave = tid >> 5;
    const int lane = tid & 31;
    const int tok0 = blockIdx.x * TM;

    v8f acc[NTILES];
    const v8f zero8 = {0.f, 0.f, 0.f, 0.f, 0.f, 0.f, 0.f, 0.f};
#pragma unroll
    for (int i = 0; i < NTILES; ++i) acc[i] = zero8;

    const int mrow   = lane & 15;
    const int ksel   = (lane >> 4) << 1;   // 0 for lanes 0-15, 2 for lanes 16-31
    const int m_base = wave * 16;
    const int col2   = (lane & 15) * 2;    // B-pair byte column base / 4

    // ---- async-stage chunk k0 into buffer b ----
    auto stage = [&](int b, int k0) {
#pragma unroll
        for (int i = 0; i < 4; ++i) {
            int idx = tid + i * 256;            // 0..1023 float4 slots of x tile
            int row = idx >> 3;
            int c4  = (idx & 7) << 2;
            async_ld_b128(lds_off(&sA[b][row * SA_STRIDE + c4]),
                          (uint64_t)(uintptr_t)(x + (size_t)(tok0 + row) * DIM + k0 + c4));
        }
        for (int i4 = tid; i4 < (TK * NCOLS) / 4; i4 += 256) {
            async_ld_b128(lds_off(&sB[b][i4 * 4]),
                          (uint64_t)(uintptr_t)(Wtp + (size_t)k0 * NCOLS + i4 * 4));
        }
    };

    int buf = 0;
    stage(0, 0);
    wait_async0();
    __syncthreads();

    for (int k0 = 0; k0 < DIM; k0 += TK) {
        if (k0 + TK < DIM) stage(buf ^ 1, k0 + TK);   // overlap with WMMAs below

        const float* __restrict__ sAc = &sA[buf][(m_base + mrow) * SA_STRIDE + ksel];
        const float* __restrict__ sBc = &sB[buf][(ksel >> 1) * PAIR_STRIDE + col2];
#pragma unroll
        for (int kk = 0; kk < TK; kk += 4) {
            v2f a = *(const v2f*)(sAc + kk);                       // one ds_load_b64
            const float* bb = sBc + (kk >> 1) * PAIR_STRIDE;
#pragma unroll
            for (int nt = 0; nt < NTILES; ++nt) {
                v2f b = *(const v2f*)(bb + nt * 32);               // one ds_load_b64
                acc[nt] = __builtin_amdgcn_wmma_f32_16x16x4_f32(
                    false, a, false, b, (short)0, acc[nt], false, false);
            }
        }
        wait_async0();
        __syncthreads();
        buf ^= 1;
    }

    // ---- spill logits to LDS: C/D layout VGPR r -> rows r / r+8 ----
    {
        const int mm = wave * 16 + ((lane >> 4) << 3);
        const int cc = lane & 15;
#pragma unroll
        for (int nt = 0; nt < NTILES; ++nt) {
#pragma unroll
            for (int r = 0; r < 8; ++r) {
                sL[(mm + r) * SL_STRIDE + nt * 16 + cc] = acc[nt][r];
            }
        }
    }
    if (tid < NEXP) sCnt[tid] = 0;
    __syncthreads();

    // ---- per-token routing: one thread per token ----
    if (tid < TM) {
        float* row = &sL[tid * SL_STRIDE];
        // group argmax (softmax monotonic; first-max-wins like jnp.argmax)
        int g = 0;
        float gmax = row[NEXP + 0];
#pragma unroll
        for (int i = 1; i < NGRP; ++i) {
            float v = row[NEXP + i];
            if (v > gmax) { gmax = v; g = i; }
        }
        const int e0 = g * 16;
        float lg[16];
        float m1 = NEGF;
#pragma unroll
        for (int i = 0; i < 16; ++i) { lg[i] = row[e0 + i]; m1 = fmaxf(m1, lg[i]); }
        float sum = 0.f;
#pragma unroll
        for (int i = 0; i < 16; ++i) { lg[i] = __expf(lg[i] - m1); sum += lg[i]; }
        const float inv = 1.0f / sum;
        // top-2 (strict > => lower index wins ties, matching lax.top_k)
        float v1 = NEGF, v2 = NEGF; int i1 = 0, i2 = 0;
#pragma unroll
        for (int i = 0; i < 16; ++i) {
            float l = row[e0 + i];
            if (l > v1)      { v2 = v1; i2 = i1; v1 = l; i1 = i; }
            else if (l > v2) { v2 = l;  i2 = i; }
        }
        const float e2 = __expf(v2 - v1);
        const float s1 = 1.0f / (1.0f + e2);
        const float s2 = e2 * s1;
        const int tg = tok0 + tid;
        out_idx[tg * 2 + 0]    = e0 + i1;
        out_idx[tg * 2 + 1]    = e0 + i2;
        out_scores[tg * 2 + 0] = s1;
        out_scores[tg * 2 + 1] = s2;
        atomicAdd(&sCnt[e0 + i1], 1);
        atomicAdd(&sCnt[e0 + i2], 1);
        // rewrite this token's LDS row with probs_full (masked experts -> exactly 0)
#pragma unroll
        for (int e = 0; e < NEXP; ++e) row[e] = 0.0f;
#pragma unroll
        for (int i = 0; i < 16; ++i) row[e0 + i] = lg[i] * inv;
    }
    __syncthreads();

    // ---- coalesced probs_full store ----
    for (int i = tid; i < TM * NEXP; i += 256) {
        out_probs[(size_t)tok0 * NEXP + i] = sL[(i >> 6) * SL_STRIDE + (i & 63)];
    }
    // ---- per-block reductions -> global accumulators ----
    if (tid < NEXP) {
        float s = 0.0f;
        for (int m = 0; m < TM; ++m) s += sL[m * SL_STRIDE + tid];
        atomicAdd(&ws_imp[tid], s);
        atomicAdd(&ws_cnt[tid], (float)sCnt[tid]);
    }
}

// ---------------- finalize importance / load ---------------------------------
__global__ void hr_final_kernel(const float* __restrict__ ws_imp,
                                const float* __restrict__ ws_cnt,
                                float* __restrict__ out_imp,
                                float* __restrict__ out_load) {
    int e = threadIdx.x;
    if (e < NEXP) {
        out_imp[e]  = ws_imp[e] * (1.0f / (float)NTOK);       // mean over B*S
        out_load[e] = ws_cnt[e] * (1.0f / (float)(NTOK * 2)); // sum(counts) = N*K
    }
}

extern "C" void kernel_launch(void* const* d_in, const int* in_sizes, int n_in,
                              void* d_out, int out_size, void* d_ws, size_t ws_size,
                              hipStream_t stream) {
    const float* x  = (const float*)d_in[0];   // [8,4096,1024]
    const float* Wg = (const float*)d_in[1];   // [4,1024]
    const float* We = (const float*)d_in[2];   // [64,1024]

    float* ws     = (float*)d_ws;
    float* wsWtp  = ws;                                  // DIM*NCOLS floats
    float* ws_imp = ws + (size_t)DIM * NCOLS;            // 64 floats
    float* ws_cnt = ws + (size_t)DIM * NCOLS + NEXP;     // 64 floats

    float* out        = (float*)d_out;
    int*   out_idx    = (int*)out;                       // [NTOK,2] int32 bits
    float* out_scores = out + (size_t)NTOK * 2;          // [NTOK,2]
    float* out_probs  = out + (size_t)NTOK * 4;          // [NTOK,64]
    float* out_imp    = out + (size_t)NTOK * 4 + (size_t)NTOK * NEXP;  // [64]
    float* out_load   = out_imp + NEXP;                  // [64]

    hr_prep_kernel<<<(DIM * NCOLS + 255) / 256, 256, 0, stream>>>(Wg, We, ws);
    hr_router_kernel<<<NTOK / TM, 256, 0, stream>>>(x, wsWtp, out_idx, out_scores,
                                                    out_probs, ws_imp, ws_cnt);
    hr_final_kernel<<<1, 64, 0, stream>>>(ws_imp, ws_cnt, out_imp, out_load);
}